// GCN_412316860738
// MI455X (gfx1250) — compile-verified
//
#include <hip/hip_runtime.h>

// Problem dims (match reference)
constexpr int NS  = 306;      // sensors
constexpr int NV  = 200000;   // vertices (divisible by 16)
constexpr int NK  = 32;       // feature dim
constexpr int NL  = 64;       // hidden dim
constexpr long long NES = 800000;
constexpr long long NEV = 1200000;

typedef __attribute__((ext_vector_type(2))) float v2f;
typedef __attribute__((ext_vector_type(8))) float v8f;

// ---------------- init: h = 0, out = broadcast(b0) ----------------
__global__ void k_init(float* __restrict__ h, float* __restrict__ out,
                       const float* __restrict__ b0) {
    long long i = (long long)blockIdx.x * blockDim.x + threadIdx.x;
    if (i < (long long)NV * NL) h[i] = 0.0f;
    if (i < (long long)NV * NK) out[i] = b0[i & (NK - 1)];
}

// ---------------- GEMM1: xw[306x64] = x[306x32] @ Ws[32x64] (WMMA f32) ----------------
// One wave per 16-row M tile (20 tiles). OOB rows handled via clamped address +
// mask multiply (flat control flow -> EXEC stays all-ones around WMMA).
__global__ void k_gemm1(const float* __restrict__ x, const float* __restrict__ Ws,
                        float* __restrict__ xw) {
    const int wave = threadIdx.x >> 5;   // M tile id, 0..19
    const int lane = threadIdx.x & 31;
    const int half = lane >> 4;          // 0: K={0,1}; 1: K={2,3}
    const int l16  = lane & 15;
    const int m0   = wave * 16;

    // Preload all A fragments for this tile (K=32 -> 8 steps of 4)
    const int row   = m0 + l16;
    const int rowc  = row < NS ? row : NS - 1;        // clamp: load always safe
    const float msk = row < NS ? 1.0f : 0.0f;         // zero OOB rows
    const float* __restrict__ xrow = x + rowc * NK;
    v2f afrag[8];
#pragma unroll
    for (int kk = 0; kk < 8; ++kk) {
        const int k0 = kk * 4 + half * 2;
        v2f t = *(const v2f*)(xrow + k0);
        afrag[kk].x = t.x * msk;
        afrag[kk].y = t.y * msk;
    }

    for (int nt = 0; nt < 4; ++nt) {     // N = 64 -> 4 tiles of 16
        v8f acc = {};
#pragma unroll
        for (int kk = 0; kk < 8; ++kk) {
            const int k0 = kk * 4 + half * 2;
            v2f b;
            b.x = Ws[(k0    ) * NL + nt * 16 + l16];
            b.y = Ws[(k0 + 1) * NL + nt * 16 + l16];
            acc = __builtin_amdgcn_wmma_f32_16x16x4_f32(
                false, afrag[kk], false, b, (short)0, acc, false, false);
        }
#pragma unroll
        for (int j = 0; j < 8; ++j) {    // D: VGPR j -> M = j (+8 for upper half-lanes)
            const int orow = m0 + j + half * 8;
            if (orow < NS) xw[orow * NL + nt * 16 + l16] = acc[j];
        }
    }
}

// ---------------- SpMM1: h[rows[e], :] += vals[e] * xw[cols[e], :]  (64 cols) ----------------
__global__ void k_spmm1(const int* __restrict__ rows, const int* __restrict__ cols,
                        const float* __restrict__ vals, const float* __restrict__ xw,
                        float* __restrict__ h) {
    const long long tid = (long long)blockIdx.x * blockDim.x + threadIdx.x;
    const long long e = tid >> 6;
    const int c = (int)(tid & 63);
    if (e >= NES) return;
    const int r  = rows[e];
    const int cl = cols[e];
    const float v = vals[e];
    atomicAdd(&h[(long long)r * NL + c], v * xw[cl * NL + c]);
}

// ---------------- GEMM2 (fused bias+ReLU on input):
//   hw[V x 32] = relu(h + b_sensor) @ W0[64 x 32]  (WMMA f32)
// One wave per 16-row M tile; A fragments loaded once (h read exactly once),
// both N-tile accumulators kept live.
__global__ void k_gemm2(const float* __restrict__ h, const float* __restrict__ bs,
                        const float* __restrict__ W0, float* __restrict__ hw) {
    const int gw = blockIdx.x * (blockDim.x >> 5) + (threadIdx.x >> 5);
    if (gw >= NV / 16) return;           // wave-uniform exit: EXEC stays all-ones
    const int lane = threadIdx.x & 31;
    const int half = lane >> 4;
    const int l16  = lane & 15;
    const int m0   = gw * 16;
    const float* __restrict__ arow = h + (long long)(m0 + l16) * NL;

    // Preload + fuse bias/ReLU: A fragments for K=64 -> 16 steps of 4
    v2f afrag[16];
#pragma unroll
    for (int kk = 0; kk < 16; ++kk) {
        const int k0 = kk * 4 + half * 2;
        v2f t  = *(const v2f*)(arow + k0);
        v2f bb = *(const v2f*)(bs + k0);
        afrag[kk].x = fmaxf(t.x + bb.x, 0.0f);
        afrag[kk].y = fmaxf(t.y + bb.y, 0.0f);
    }

    v8f acc0 = {};
    v8f acc1 = {};
#pragma unroll
    for (int kk = 0; kk < 16; ++kk) {
        const int k0 = kk * 4 + half * 2;
        v2f b0f, b1f;
        b0f.x = W0[(k0    ) * NK +      l16];
        b0f.y = W0[(k0 + 1) * NK +      l16];
        b1f.x = W0[(k0    ) * NK + 16 + l16];
        b1f.y = W0[(k0 + 1) * NK + 16 + l16];
        acc0 = __builtin_amdgcn_wmma_f32_16x16x4_f32(
            false, afrag[kk], false, b0f, (short)0, acc0, false, false);
        acc1 = __builtin_amdgcn_wmma_f32_16x16x4_f32(
            false, afrag[kk], false, b1f, (short)0, acc1, false, false);
    }
#pragma unroll
    for (int j = 0; j < 8; ++j) {
        const int orow = m0 + j + half * 8;
        hw[(long long)orow * NK +      l16] = acc0[j];
        hw[(long long)orow * NK + 16 + l16] = acc1[j];
    }
}

// ---------------- SpMM2: out[rows[e], :] += vals[e] * hw[cols[e], :]  (32 cols) ----------------
__global__ void k_spmm2(const int* __restrict__ rows, const int* __restrict__ cols,
                        const float* __restrict__ vals, const float* __restrict__ hw,
                        float* __restrict__ out) {
    const long long tid = (long long)blockIdx.x * blockDim.x + threadIdx.x;
    const long long e = tid >> 5;
    const int c = (int)(tid & 31);
    if (e >= NEV) return;
    const int r  = rows[e];
    const int cl = cols[e];
    const float v = vals[e];
    atomicAdd(&out[(long long)r * NK + c], v * hw[(long long)cl * NK + c]);
}

extern "C" void kernel_launch(void* const* d_in, const int* in_sizes, int n_in,
                              void* d_out, int out_size, void* d_ws, size_t ws_size,
                              hipStream_t stream) {
    const float* x    = (const float*)d_in[0];
    const int*   srow = (const int*)  d_in[1];
    const int*   scol = (const int*)  d_in[2];
    const float* sval = (const float*)d_in[3];
    const int*   vrow = (const int*)  d_in[4];
    const int*   vcol = (const int*)  d_in[5];
    const float* vval = (const float*)d_in[6];
    const float* Ws   = (const float*)d_in[7];
    const float* bs   = (const float*)d_in[8];
    const float* W0   = (const float*)d_in[9];
    const float* b0   = (const float*)d_in[10];
    float* out = (float*)d_out;

    // Workspace layout (floats): xw (padded to 32768) | h (V*64) | hw (V*32)
    float* ws = (float*)d_ws;
    float* xw = ws;
    float* h  = ws + 32768;
    float* hw = h + (size_t)NV * NL;

    const long long nInit = (long long)NV * NL;           // 12.8M covers both h and out
    k_init<<<(unsigned)((nInit + 255) / 256), 256, 0, stream>>>(h, out, b0);

    k_gemm1<<<1, 640, 0, stream>>>(x, Ws, xw);            // 20 waves, one per M tile

    const long long n1 = NES * 64;
    k_spmm1<<<(unsigned)((n1 + 255) / 256), 256, 0, stream>>>(srow, scol, sval, xw, h);

    const int tilesM = NV / 16;                           // 12500
    const int wavesPerBlock = 8;                          // 256 threads
    const int blocks = (tilesM + wavesPerBlock - 1) / wavesPerBlock;
    k_gemm2<<<blocks, 32 * wavesPerBlock, 0, stream>>>(h, bs, W0, hw);

    const long long n2 = NEV * 32;
    k_spmm2<<<(unsigned)((n2 + 255) / 256), 256, 0, stream>>>(vrow, vcol, vval, hw, out);
}